// Routing_78855599554886
// MI455X (gfx1250) — compile-verified
//
#include <hip/hip_runtime.h>
#include <hip/hip_bf16.h>

typedef __attribute__((ext_vector_type(2))) float v2f;
typedef __attribute__((ext_vector_type(8))) float v8f;
typedef __attribute__((ext_vector_type(4))) int v4i;

constexpr int NB = 512;    // batch
constexpr int NI = 315;    // n_inp (i)
constexpr int DD = 1024;   // d_inp (d)
constexpr int NJ = 4;      // n_out (j)
constexpr int TI = 64;     // i-rows per block (4 waves x 16)
constexpr int KC = 64;     // K (d) chunk staged in LDS
constexpr int XS = 68;     // padded LDS row stride
constexpr int BPS = 68;    // bpad row stride
constexpr int NTILE = 5;   // ceil(315/64)
constexpr int NCH = DD / KC;

#if __has_builtin(__builtin_amdgcn_global_load_async_to_lds_b128)
#define USE_ASYNC 1
#else
#define USE_ASYNC 0
#endif

#if USE_ASYNC
typedef __attribute__((address_space(1))) v4i* as1_v4i;
typedef __attribute__((address_space(3))) v4i* as3_v4i;
#endif

__device__ __forceinline__ void copy16_to_lds(const float* g, float* l)
{
#if USE_ASYNC
    __builtin_amdgcn_global_load_async_to_lds_b128((as1_v4i)g, (as3_v4i)l, 0, 0);
#else
    *(float4*)l = *(const float4*)g;
#endif
}

template <int N>
__device__ __forceinline__ void wait_async()
{
#if USE_ASYNC
#if __has_builtin(__builtin_amdgcn_s_wait_asynccnt)
    __builtin_amdgcn_s_wait_asynccnt(N);
#else
    asm volatile("s_wait_asynccnt %0" ::"n"(N) : "memory");
#endif
#endif
}

// ---------------------------------------------------------------------------
// Pass A: per (b, i-tile): a_inp -> f, c[b,i,j], partial x_out0 contributions.
// WMMA chain 1: C_a = X_tile (16xK) * W_A_tile^T (Kx16); diag = <x_i, W_A[i]>
// WMMA chain 2: C_c = X_tile * bpad (cols 0..3 = W_F1[j]*W_F2, cols 4..15 = 0)
// Double-buffered LDS staging via async global->LDS copies.
// ---------------------------------------------------------------------------
__global__ __launch_bounds__(128)
void passA_kernel(const float* __restrict__ x, const float* __restrict__ WA,
                  const float* __restrict__ BA, const float* __restrict__ WF1,
                  const float* __restrict__ WF2, const float* __restrict__ BF2,
                  const float* __restrict__ bu, const float* __restrict__ bi,
                  float* __restrict__ ws_f, float* __restrict__ ws_c,
                  float* __restrict__ ws_p0)
{
    __shared__ float ldsX[2][TI * XS];
    __shared__ float ldsW[2][TI * XS];
    __shared__ float bpad[2][16 * BPS];
    __shared__ float ldsCA[TI * 17];
    __shared__ float ldsCC[TI * 17];
    __shared__ float red[TI * NJ];

    const int tid = threadIdx.x;
    const int wv  = tid >> 5;
    const int ln  = tid & 31;
    const int b   = blockIdx.y;
    const int i0  = blockIdx.x * TI;
    const int m   = ln & 15;
    const int h   = ln >> 4;
    const float invs = 1.0f / sqrtf((float)NI);

    // rows 4..15 of both bpad buffers are permanent zeros
    for (int idx = tid; idx < 12 * BPS; idx += 128) {
        bpad[0][4 * BPS + idx] = 0.f;
        bpad[1][4 * BPS + idx] = 0.f;
    }

    auto issue = [&](int ch, int p) {
        const int k0 = ch * KC;
        #pragma unroll
        for (int q = 0; q < 8; ++q) {
            int slot = q * 128 + tid;
            int r    = slot >> 4;
            int c4   = (slot & 15) * 4;
            int ic   = i0 + r;
            if (ic > NI - 1) ic = NI - 1;  // clamp: garbage rows feed discarded C rows
            copy16_to_lds(x + ((size_t)b * NI + ic) * DD + k0 + c4,
                          &ldsX[p][r * XS + c4]);
            copy16_to_lds(WA + (size_t)ic * DD + k0 + c4,
                          &ldsW[p][r * XS + c4]);
        }
        #pragma unroll
        for (int q = 0; q < 2; ++q) {
            int e = q * 128 + tid;     // 256 floats: rows 0..3 of bpad
            int j = e >> 6, d = e & 63;
            bpad[p][j * BPS + d] = WF1[j * DD + k0 + d] * WF2[k0 + d];
        }
    };

    v8f ca = {};
    v8f cc = {};
    issue(0, 0);

    #pragma unroll 1
    for (int c = 0; c < NCH; ++c) {
        const int p = c & 1;
        if (c + 1 < NCH) {
            issue(c + 1, p ^ 1);
            wait_async<16>();   // 16 async b128 issues/wave/stage -> stage c landed
        } else {
            wait_async<0>();
        }
        __syncthreads();
        const float* Xrow = &ldsX[p][(wv * 16 + m) * XS];
        const float* Wrow = &ldsW[p][(wv * 16 + m) * XS];
        const float* Brow = &bpad[p][m * BPS];
        #pragma unroll
        for (int kk = 0; kk < KC; kk += 4) {
            v2f a, bwa, bc;
            a.x   = Xrow[kk + 2 * h];
            a.y   = Xrow[kk + 2 * h + 1];
            bwa.x = Wrow[kk + 2 * h];
            bwa.y = Wrow[kk + 2 * h + 1];
            bc.x  = Brow[kk + 2 * h];
            bc.y  = Brow[kk + 2 * h + 1];
            ca = __builtin_amdgcn_wmma_f32_16x16x4_f32(false, a, false, bwa, (short)0, ca, false, false);
            cc = __builtin_amdgcn_wmma_f32_16x16x4_f32(false, a, false, bc,  (short)0, cc, false, false);
        }
        __syncthreads();
    }

    // dump accumulators (C layout: vgpr r, lane l -> M=r+8*(l>>4), N=l&15)
    #pragma unroll
    for (int r = 0; r < 8; ++r) {
        int mm = r + 8 * h;
        ldsCA[(wv * 16 + mm) * 17 + m] = ca[r];
        ldsCC[(wv * 16 + mm) * 17 + m] = cc[r];
    }
    for (int idx = tid; idx < TI * NJ; idx += 128) red[idx] = 0.f;
    __syncthreads();

    if (tid < TI) {
        int i = i0 + tid;
        if (i < NI) {
            float dotA = ldsCA[tid * 17 + (tid & 15)];  // diagonal element
            float aval = invs * dotA + BA[i];
            float f = 1.0f / (1.0f + __expf(-aval));
            ws_f[b * NI + i] = f;
            #pragma unroll
            for (int j = 0; j < NJ; ++j) {
                float cj = invs * ldsCC[tid * 17 + j];
                ws_c[((size_t)(b * NI + i)) * NJ + j] = cj;
                float phi = f * (0.25f * bu[i * NJ + j] - 0.75f * bi[i * NJ + j]);
                red[tid * NJ + j] = phi * (cj + BF2[j]);
            }
        }
    }
    __syncthreads();
    if (tid < NJ) {  // deterministic fixed-order tile reduction
        float s = 0.f;
        for (int t = 0; t < TI; ++t) s += red[t * NJ + tid];
        ws_p0[(b * NTILE + blockIdx.x) * NJ + tid] = s;
    }
}

// ---------------------------------------------------------------------------
// Pass B: bpad rows 0..3 = pred[b,j,chunk] built on the fly; WMMA C_t = X*pred^T.
// ---------------------------------------------------------------------------
__global__ __launch_bounds__(128)
void passB_kernel(const float* __restrict__ x, const float* __restrict__ WG1,
                  const float* __restrict__ WG2, const float* __restrict__ BG2,
                  const float* __restrict__ WS, const float* __restrict__ BS,
                  const float* __restrict__ BF2,
                  const float* __restrict__ bu, const float* __restrict__ bi,
                  const float* __restrict__ ws_f, const float* __restrict__ ws_c,
                  const float* __restrict__ ws_xo0, float* __restrict__ ws_p1)
{
    __shared__ float ldsX[2][TI * XS];
    __shared__ float bpad[2][16 * BPS];
    __shared__ float ldsCT[TI * 17];
    __shared__ float red[TI * NJ];

    const int tid = threadIdx.x;
    const int wv  = tid >> 5;
    const int ln  = tid & 31;
    const int b   = blockIdx.y;
    const int i0  = blockIdx.x * TI;
    const int m   = ln & 15;
    const int h   = ln >> 4;

    for (int idx = tid; idx < 12 * BPS; idx += 128) {
        bpad[0][4 * BPS + idx] = 0.f;
        bpad[1][4 * BPS + idx] = 0.f;
    }

    auto issue = [&](int ch, int p) {
        const int k0 = ch * KC;
        #pragma unroll
        for (int q = 0; q < 8; ++q) {
            int slot = q * 128 + tid;
            int r    = slot >> 4;
            int c4   = (slot & 15) * 4;
            int ic   = i0 + r;
            if (ic > NI - 1) ic = NI - 1;
            copy16_to_lds(x + ((size_t)b * NI + ic) * DD + k0 + c4,
                          &ldsX[p][r * XS + c4]);
        }
        #pragma unroll
        for (int q = 0; q < 2; ++q) {
            int e = q * 128 + tid;
            int j = e >> 6, d = e & 63;
            int gd = k0 + d;
            // pred[j][gd] = x_out0[b,j] * W_G1[0,gd] * W_G2[j,gd] + B_G2[j,gd]
            bpad[p][j * BPS + d] =
                ws_xo0[b * NJ + j] * WG1[gd] * WG2[j * DD + gd] + BG2[j * DD + gd];
        }
    };

    v8f ct = {};
    issue(0, 0);

    #pragma unroll 1
    for (int c = 0; c < NCH; ++c) {
        const int p = c & 1;
        if (c + 1 < NCH) {
            issue(c + 1, p ^ 1);
            wait_async<8>();    // 8 async b128 issues/wave/stage
        } else {
            wait_async<0>();
        }
        __syncthreads();
        const float* Xrow = &ldsX[p][(wv * 16 + m) * XS];
        const float* Brow = &bpad[p][m * BPS];
        #pragma unroll
        for (int kk = 0; kk < KC; kk += 4) {
            v2f a, bp;
            a.x  = Xrow[kk + 2 * h];
            a.y  = Xrow[kk + 2 * h + 1];
            bp.x = Brow[kk + 2 * h];
            bp.y = Brow[kk + 2 * h + 1];
            ct = __builtin_amdgcn_wmma_f32_16x16x4_f32(false, a, false, bp, (short)0, ct, false, false);
        }
        __syncthreads();
    }

    #pragma unroll
    for (int r = 0; r < 8; ++r) {
        int mm = r + 8 * h;
        ldsCT[(wv * 16 + mm) * 17 + m] = ct[r];
    }
    for (int idx = tid; idx < TI * NJ; idx += 128) red[idx] = 0.f;
    __syncthreads();

    if (tid < TI) {
        int i = i0 + tid;
        if (i < NI) {
            float S[NJ];
            #pragma unroll
            for (int j = 0; j < NJ; ++j) {
                float t = ldsCT[tid * 17 + j];
                float z = t * WS[i * NJ + j] + BS[i * NJ + j];
                // log_sigmoid(z) = min(z,0) - log1p(exp(-|z|))
                S[j] = fminf(z, 0.f) - log1pf(__expf(-fabsf(z)));
            }
            float mx = fmaxf(fmaxf(S[0], S[1]), fmaxf(S[2], S[3]));
            float e[NJ], sum = 0.f;
            #pragma unroll
            for (int j = 0; j < NJ; ++j) { e[j] = __expf(S[j] - mx); sum += e[j]; }
            float rs = 1.0f / sum;
            float f = ws_f[b * NI + i];
            #pragma unroll
            for (int j = 0; j < NJ; ++j) {
                float R   = e[j] * rs;
                float bus = bu[i * NJ + j];
                float bis = bi[i * NJ + j];
                float phi = f * (R * (bus + bis) - bis);
                red[tid * NJ + j] =
                    phi * (ws_c[((size_t)(b * NI + i)) * NJ + j] + BF2[j]);
            }
        }
    }
    __syncthreads();
    if (tid < NJ) {
        float s = 0.f;
        for (int t = 0; t < TI; ++t) s += red[t * NJ + tid];
        ws_p1[(b * NTILE + blockIdx.x) * NJ + tid] = s;
    }
}

// Deterministic reduction of per-tile partials [NB][NTILE][NJ] -> [NB][NJ]
__global__ void reduce_kernel(const float* __restrict__ part, float* __restrict__ out)
{
    int idx = blockIdx.x * blockDim.x + threadIdx.x;
    if (idx < NB * NJ) {
        int b = idx >> 2, j = idx & 3;
        float s = 0.f;
        for (int t = 0; t < NTILE; ++t) s += part[(b * NTILE + t) * NJ + j];
        out[idx] = s;
    }
}

extern "C" void kernel_launch(void* const* d_in, const int* in_sizes, int n_in,
                              void* d_out, int out_size, void* d_ws, size_t ws_size,
                              hipStream_t stream)
{
    const float* x   = (const float*)d_in[0];   // [512,315,1024]
    const float* WA  = (const float*)d_in[1];   // [315,1024]
    const float* BA  = (const float*)d_in[2];   // [315]
    const float* WF1 = (const float*)d_in[3];   // [4,1024]
    const float* WF2 = (const float*)d_in[4];   // [1024,1]
    const float* BF2 = (const float*)d_in[5];   // [4,1]
    const float* WG1 = (const float*)d_in[6];   // [1,1024]
    const float* WG2 = (const float*)d_in[7];   // [4,1024]
    const float* BG2 = (const float*)d_in[8];   // [4,1024]
    const float* WS  = (const float*)d_in[9];   // [315,4]
    const float* BS  = (const float*)d_in[10];  // [315,4]
    const float* bu  = (const float*)d_in[11];  // [315,4]
    const float* bi  = (const float*)d_in[12];  // [315,4]
    float* out = (float*)d_out;                 // [512,4]

    float* ws     = (float*)d_ws;
    float* ws_f   = ws;                                    // 512*315
    float* ws_c   = ws_f + (size_t)NB * NI;                // 512*315*4
    float* ws_p0  = ws_c + (size_t)NB * NI * NJ;           // 512*5*4
    float* ws_xo0 = ws_p0 + (size_t)NB * NTILE * NJ;       // 512*4
    float* ws_p1  = ws_xo0 + (size_t)NB * NJ;              // 512*5*4

    dim3 grid(NTILE, NB);
    passA_kernel<<<grid, 128, 0, stream>>>(x, WA, BA, WF1, WF2, BF2, bu, bi,
                                           ws_f, ws_c, ws_p0);
    reduce_kernel<<<(NB * NJ + 255) / 256, 256, 0, stream>>>(ws_p0, ws_xo0);
    passB_kernel<<<grid, 128, 0, stream>>>(x, WG1, WG2, BG2, WS, BS, BF2, bu, bi,
                                           ws_f, ws_c, ws_xo0, ws_p1);
    reduce_kernel<<<(NB * NJ + 255) / 256, 256, 0, stream>>>(ws_p1, out);
}